// Net_cell_79714593014344
// MI455X (gfx1250) — compile-verified
//
#include <hip/hip_runtime.h>
#include <hip/hip_bf16.h>

typedef __attribute__((ext_vector_type(2))) float v2f;
typedef __attribute__((ext_vector_type(8))) float v8f;

#define TB 256

// ---------------------------------------------------------------------------
// Precompute: t = wo @ hconv_lin (512), u = wv @ t (64),
//             c = 2*(bv.t) + bo.hconv_lin
// ---------------------------------------------------------------------------
__global__ void k_pre(const float* __restrict__ wv, const float* __restrict__ bv,
                      const float* __restrict__ wo, const float* __restrict__ bo,
                      const float* __restrict__ hlin,
                      float* __restrict__ u, float* __restrict__ cb) {
  __shared__ float ts[512];
  __shared__ float cpart[2];
  int k = threadIdx.x;
  float s = 0.f;
  for (int j = 0; j < 64; ++j) s += wo[k * 64 + j] * hlin[j];
  ts[k] = s;
  __syncthreads();
  if (k < 64) {
    float su = 0.f;
    for (int q = 0; q < 512; ++q) su += wv[k * 512 + q] * ts[q];
    u[k] = su;
  }
  if (k == 64) {
    float c1 = 0.f;
    for (int q = 0; q < 512; ++q) c1 += 2.f * bv[q] * ts[q];
    cpart[0] = c1;
  }
  if (k == 65) {
    float c2 = 0.f;
    for (int j = 0; j < 64; ++j) c2 += bo[j] * hlin[j];
    cpart[1] = c2;
  }
  __syncthreads();
  if (k == 0) cb[0] = cpart[0] + cpart[1];
}

__global__ void k_zero(float* __restrict__ p, long n) {
  long i = (long)blockIdx.x * blockDim.x + threadIdx.x;
  if (i < n) p[i] = 0.f;
}

// ---------------------------------------------------------------------------
// Node phase with f32 WMMA 16x16x4:
//   xl[n] = (emb[n]+hattr[n]) . u + c       -> D column 0
//   el[n] = hattr[n] . hconv_lin            -> D column 1
// One wave = one 16-row tile; K = 64 = 16 steps of 4; two chained WMMAs per
// step (A=emb with B1=[u|0...], A=hattr with B2=[u|hlin|0...]).
// B operands are loaded branch-free: each lane picks its source pointer once
// (u / hconv_lin / zero buffer), keeping control flow uniform in the loop.
// ---------------------------------------------------------------------------
__global__ void k_nodefeat(const float* __restrict__ emb, const float* __restrict__ hat,
                           const float* __restrict__ hlin, const float* __restrict__ u,
                           const float* __restrict__ zbuf, const float* __restrict__ cb,
                           float* __restrict__ xl, float* __restrict__ el, int N) {
  int lane = threadIdx.x & 31;
  int wave = threadIdx.x >> 5;
  int tile = blockIdx.x * (blockDim.x >> 5) + wave;
  long base = (long)tile * 16;
  if (base >= N) return;                 // uniform per wave
  const float cconst = cb[0];
  const int lmod = lane & 15;
  const int half = lane >> 4;            // 0: K pair {0,1}, 1: K pair {2,3}

  if (base + 16 <= (long)N) {
    // per-lane B source pointers (chosen once; loop body is branch-free)
    const float* bs1 = (lmod == 0) ? u : zbuf;                          // B1 col0=u
    const float* bs2 = (lmod == 0) ? u : ((lmod == 1) ? hlin : zbuf);   // B2 col0=u,col1=hlin
    const float* rowe = emb + (base + lmod) * 64 + 2 * half;
    const float* rowh = hat + (base + lmod) * 64 + 2 * half;
    const float* b1p  = bs1 + 2 * half;
    const float* b2p  = bs2 + 2 * half;
    v8f acc = {};
#pragma unroll
    for (int s = 0; s < 16; ++s) {
      const int off = 4 * s;             // +2*half folded into the pointers
      v2f a1 = *(const v2f*)(rowe + off);      // A = emb tile
      v2f a2 = *(const v2f*)(rowh + off);      // A = hattr tile
      v2f b1 = *(const v2f*)(b1p + off);
      v2f b2 = *(const v2f*)(b2p + off);
      acc = __builtin_amdgcn_wmma_f32_16x16x4_f32(false, a1, false, b1,
                                                  (short)0, acc, false, false);
      acc = __builtin_amdgcn_wmma_f32_16x16x4_f32(false, a2, false, b2,
                                                  (short)0, acc, false, false);
    }
    // D layout: VGPR j, lanes 0-15 -> D[j][lane]; lanes 16-31 -> D[j+8][lane-16]
#pragma unroll
    for (int j = 0; j < 8; ++j) {
      int row = (int)base + j + 8 * half;
      float v = acc[j];
      if (lmod == 0) xl[row] = v + cconst;
      if (lmod == 1) el[row] = v;
    }
  } else {
    // scalar tail for a partial tile (N not multiple of 16)
    for (long r = base + lane; r < (long)N; r += 32) {
      float sx = 0.f, se = 0.f;
      for (int kk = 0; kk < 64; ++kk) {
        float hv = hat[r * 64 + kk];
        sx += (emb[r * 64 + kk] + hv) * u[kk];
        se += hv * hlin[kk];
      }
      xl[r] = sx + cconst;
      el[r] = se;
    }
  }
}

// ---------------------------------------------------------------------------
// Edge pass 1: degree counts + raw attention logit + segment max (monotone
// uint encoding so atomicMax works for signed floats).
// ---------------------------------------------------------------------------
__device__ __forceinline__ unsigned f2mono(float f) {
  unsigned b = __float_as_uint(f);
  return (b & 0x80000000u) ? ~b : (b | 0x80000000u);
}
__device__ __forceinline__ float mono2f(unsigned u) {
  return (u & 0x80000000u) ? __uint_as_float(u ^ 0x80000000u) : __uint_as_float(~u);
}

__global__ void k_edge1(const int* __restrict__ i0p, const int* __restrict__ i1p,
                        const float* __restrict__ xl, const float* __restrict__ el,
                        const float* __restrict__ att,
                        float* __restrict__ araw, unsigned* __restrict__ amaxu,
                        int* __restrict__ Bdeg, int* __restrict__ Ddeg, int nnz) {
  int e = blockIdx.x * blockDim.x + threadIdx.x;
  if (e >= nnz) return;
  int i0 = i0p[e], i1 = i1p[e];
  atomicAdd(&Ddeg[i0], 1);
  atomicAdd(&Bdeg[i1], 1);
  float a = xl[i0] * att[0] + el[i1] * att[1];
  a = a > 0.f ? a : 0.2f * a;            // leaky_relu(0.2)
  araw[e] = a;
  atomicMax(&amaxu[i1], f2mono(a));
}

__global__ void k_inv(const int* __restrict__ Bdeg, const int* __restrict__ Ddeg,
                      float* __restrict__ Binv, float* __restrict__ Dinv, int n) {
  int i = blockIdx.x * blockDim.x + threadIdx.x;
  if (i >= n) return;
  Binv[i] = Bdeg[i] > 0 ? 1.f / (float)Bdeg[i] : 0.f;
  Dinv[i] = Ddeg[i] > 0 ? 1.f / (float)Ddeg[i] : 0.f;
}

// Edge pass 2: aexp = exp(a - segmax); segment sum
__global__ void k_exp(const int* __restrict__ i1p, float* __restrict__ aexp,
                      const unsigned* __restrict__ amaxu, float* __restrict__ asum,
                      int nnz) {
  int e = blockIdx.x * blockDim.x + threadIdx.x;
  if (e >= nnz) return;
  int i1 = i1p[e];
  float ex = expf(aexp[e] - mono2f(amaxu[i1]));
  aexp[e] = ex;
  atomicAdd(&asum[i1], ex);
}

__global__ void k_s1(const float* __restrict__ Binv, const float* __restrict__ asum,
                     float* __restrict__ s1, int n) {
  int i = blockIdx.x * blockDim.x + threadIdx.x;
  if (i >= n) return;
  s1[i] = asum[i] > 0.f ? Binv[i] / asum[i] : 0.f;
}

// Propagate 1, hop a: ef[i1] += aexp * (Binv/asum)[i1] * xl[i0]
__global__ void k_prop1a(const int* __restrict__ i0p, const int* __restrict__ i1p,
                         const float* __restrict__ aexp, const float* __restrict__ s1,
                         const float* __restrict__ xl, float* __restrict__ ef, int nnz) {
  int e = blockIdx.x * blockDim.x + threadIdx.x;
  if (e >= nnz) return;
  int i0 = i0p[e], i1 = i1p[e];
  atomicAdd(&ef[i1], aexp[e] * s1[i1] * xl[i0]);
}

__global__ void k_efs(const float* __restrict__ ef, const float* __restrict__ asum,
                      float* __restrict__ efs, int n) {
  int i = blockIdx.x * blockDim.x + threadIdx.x;
  if (i >= n) return;
  efs[i] = asum[i] > 0.f ? ef[i] / asum[i] : 0.f;
}

// Propagate 1, hop b: h[i0] += aexp * (ef/asum)[i1] * Dinv[i0]
__global__ void k_prop1b(const int* __restrict__ i0p, const int* __restrict__ i1p,
                         const float* __restrict__ aexp, const float* __restrict__ efs,
                         const float* __restrict__ Dinv, float* __restrict__ h, int nnz) {
  int e = blockIdx.x * blockDim.x + threadIdx.x;
  if (e >= nnz) return;
  int i0 = i0p[e], i1 = i1p[e];
  atomicAdd(&h[i0], aexp[e] * efs[i1] * Dinv[i0]);
}

__global__ void k_hl(const float* __restrict__ h, const float* __restrict__ hcb,
                     const float* __restrict__ hlw, float* __restrict__ hl, int n) {
  int i = blockIdx.x * blockDim.x + threadIdx.x;
  if (i >= n) return;
  hl[i] = (h[i] + hcb[0]) * hlw[0];
}

// Propagate 2, hop a: ef3[i1] += Binv[i1] * hl[i0]
__global__ void k_prop2a(const int* __restrict__ i0p, const int* __restrict__ i1p,
                         const float* __restrict__ Binv, const float* __restrict__ hl,
                         float* __restrict__ ef3, int nnz) {
  int e = blockIdx.x * blockDim.x + threadIdx.x;
  if (e >= nnz) return;
  atomicAdd(&ef3[i1p[e]], Binv[i1p[e]] * hl[i0p[e]]);
}

// Propagate 2, hop b: outacc[i0] += Dinv[i0] * ef3[i1]
__global__ void k_prop2b(const int* __restrict__ i0p, const int* __restrict__ i1p,
                         const float* __restrict__ Dinv, const float* __restrict__ ef3,
                         float* __restrict__ outacc, int nnz) {
  int e = blockIdx.x * blockDim.x + threadIdx.x;
  if (e >= nnz) return;
  atomicAdd(&outacc[i0p[e]], Dinv[i0p[e]] * ef3[i1p[e]]);
}

// Final: r = relu(outacc + head_bias); out[n,c] = r*out_w[c] + out_b[c]
__global__ void k_out(const float* __restrict__ outacc, const float* __restrict__ hb,
                      const float* __restrict__ ow, const float* __restrict__ ob,
                      float* __restrict__ out, int n, int C) {
  int i = blockIdx.x * blockDim.x + threadIdx.x;
  if (i >= n) return;
  float r = outacc[i] + hb[0];
  r = r > 0.f ? r : 0.f;
  for (int c = 0; c < C; ++c) out[i * C + c] = r * ow[c] + ob[c];
}

// ---------------------------------------------------------------------------
extern "C" void kernel_launch(void* const* d_in, const int* in_sizes, int n_in,
                              void* d_out, int out_size, void* d_ws, size_t ws_size,
                              hipStream_t stream) {
  const float* emb  = (const float*)d_in[0];
  const int*   ei   = (const int*)d_in[1];
  const float* hat  = (const float*)d_in[2];
  const float* wv   = (const float*)d_in[7];
  const float* bv   = (const float*)d_in[8];
  const float* wo   = (const float*)d_in[9];
  const float* bo   = (const float*)d_in[10];
  const float* hlin = (const float*)d_in[11];
  const float* att  = (const float*)d_in[12];
  const float* hcb  = (const float*)d_in[13];
  const float* hlw  = (const float*)d_in[14];
  const float* hb   = (const float*)d_in[15];
  const float* ow   = (const float*)d_in[16];
  const float* ob   = (const float*)d_in[17];

  const int N   = in_sizes[0] / 64;
  const int NNZ = in_sizes[1] / 2;
  const int C   = in_sizes[16];
  const int*   i0p = ei;
  const int*   i1p = ei + NNZ;

  float* ws = (float*)d_ws;
  size_t o = 0;
  float* aexp = ws + o; o += (size_t)NNZ;   // raw logits, then exp(logit-max)
  float* xl   = ws + o; o += N;
  float* el   = ws + o; o += N;
  float* Binv = ws + o; o += N;
  float* Dinv = ws + o; o += N;
  float* s1   = ws + o; o += N;
  float* efs  = ws + o; o += N;
  float* hl   = ws + o; o += N;
  float* u    = ws + o; o += 64;
  float* cb   = ws + o; o += 16;
  float* zreg = ws + o;                     // contiguous zero-init region:
  unsigned* amaxu = (unsigned*)(zreg);      //  [amaxu][asum][Bdeg][Ddeg][ef][h][ef3][outacc][zbuf]
  float*    asum  = zreg + (size_t)N;
  int*      Bdeg  = (int*)(zreg + 2 * (size_t)N);
  int*      Ddeg  = (int*)(zreg + 3 * (size_t)N);
  float*    ef    = zreg + 4 * (size_t)N;
  float*    h     = zreg + 5 * (size_t)N;
  float*    ef3   = zreg + 6 * (size_t)N;
  float*    oacc  = zreg + 7 * (size_t)N;
  float*    zbuf  = zreg + 8 * (size_t)N;   // 64-float zero source for inactive B lanes

  float* out = (float*)d_out;

  const int gN = (N + TB - 1) / TB;
  const int gE = (NNZ + TB - 1) / TB;
  const long zn = 8L * N + 64;
  const int gZ = (int)((zn + TB - 1) / TB);
  const int tiles = (N + 15) / 16;
  const int gT = (tiles + 7) / 8;           // 8 waves (16-row tiles) per 256-thread block

  k_pre<<<1, 512, 0, stream>>>(wv, bv, wo, bo, hlin, u, cb);
  k_zero<<<gZ, TB, 0, stream>>>(zreg, zn);
  k_nodefeat<<<gT, TB, 0, stream>>>(emb, hat, hlin, u, zbuf, cb, xl, el, N);
  k_edge1<<<gE, TB, 0, stream>>>(i0p, i1p, xl, el, att, aexp, amaxu, Bdeg, Ddeg, NNZ);
  k_inv<<<gN, TB, 0, stream>>>(Bdeg, Ddeg, Binv, Dinv, N);
  k_exp<<<gE, TB, 0, stream>>>(i1p, aexp, amaxu, asum, NNZ);
  k_s1<<<gN, TB, 0, stream>>>(Binv, asum, s1, N);
  k_prop1a<<<gE, TB, 0, stream>>>(i0p, i1p, aexp, s1, xl, ef, NNZ);
  k_efs<<<gN, TB, 0, stream>>>(ef, asum, efs, N);
  k_prop1b<<<gE, TB, 0, stream>>>(i0p, i1p, aexp, efs, Dinv, h, NNZ);
  k_hl<<<gN, TB, 0, stream>>>(h, hcb, hlw, hl, N);
  k_prop2a<<<gE, TB, 0, stream>>>(i0p, i1p, Binv, hl, ef3, NNZ);
  k_prop2b<<<gE, TB, 0, stream>>>(i0p, i1p, Dinv, ef3, oacc, NNZ);
  k_out<<<gN, TB, 0, stream>>>(oacc, hb, ow, ob, out, N, C);
}